// SAGEConv_34050500722692
// MI455X (gfx1250) — compile-verified
//
#include <hip/hip_runtime.h>

// ---------------------------------------------------------------------------
// SAGEConv forward for MI455X (gfx1250, wave32):
//   agg = segment_sum(x[col] * ew, row)          (L2-resident scatter SpMM)
//   out = agg @ w_l^T + b_l + x @ w_r^T          (fused K=256 f32 WMMA GEMM)
// ---------------------------------------------------------------------------

typedef float v2f __attribute__((ext_vector_type(2)));
typedef float v8f __attribute__((ext_vector_type(8)));

#define D_FEAT 128
#define K_TOT  256          // concatenated [agg | x] reduction dim
#define AT_STRIDE 17        // 16 rows + 1 pad  -> conflict-free LDS
#define WT_STRIDE 129       // 128 cols + 1 pad -> conflict-free LDS

// -------------------- kernel 1: zero the aggregation buffer ----------------
__global__ void __launch_bounds__(256)
sage_zero_kernel(float4* __restrict__ p, int n4) {
    int i = blockIdx.x * 256 + threadIdx.x;
    if (i < n4) p[i] = make_float4(0.f, 0.f, 0.f, 0.f);
}

// -------------------- kernel 2: edge-parallel weighted scatter-add ---------
// One wave32 per edge: lane L loads float4 L of the 128-float source row
// (coalesced 512B), scales by edge weight, and does 4 native f32 atomics
// into agg (L2-resident: agg = 51.2 MB << 192 MB L2).
__global__ void __launch_bounds__(256)
sage_scatter_kernel(const float* __restrict__ x,
                    const int*   __restrict__ row,
                    const int*   __restrict__ col,
                    const float* __restrict__ ew,
                    float*       __restrict__ agg,
                    int E) {
    int tid  = blockIdx.x * 256 + threadIdx.x;
    int e    = tid >> 5;          // wave-uniform edge id
    int lane = tid & 31;
    if (e >= E) return;

    int   c = col[e];
    int   r = row[e];
    float w = ew[e];

    const float4* xs = (const float4*)(x + (size_t)c * D_FEAT);
    float4 v = xs[lane];

    float* a = agg + (size_t)r * D_FEAT + lane * 4;
    __hip_atomic_fetch_add(a + 0, v.x * w, __ATOMIC_RELAXED, __HIP_MEMORY_SCOPE_AGENT);
    __hip_atomic_fetch_add(a + 1, v.y * w, __ATOMIC_RELAXED, __HIP_MEMORY_SCOPE_AGENT);
    __hip_atomic_fetch_add(a + 2, v.z * w, __ATOMIC_RELAXED, __HIP_MEMORY_SCOPE_AGENT);
    __hip_atomic_fetch_add(a + 3, v.w * w, __ATOMIC_RELAXED, __HIP_MEMORY_SCOPE_AGENT);
}

// -------------------- kernel 3: fused dual-GEMM epilogue via f32 WMMA ------
// out[m, n] = sum_k A[m,k] * W[n,k] + b_l[n],  A = [agg | x] (K=256),
// W = [w_l | w_r]. Block = 8 waves; block owns one 16-row M tile
// (100000 = 6250 * 16 exactly), wave i owns N tile i (8 * 16 = 128 cols).
// A tile + transposed weights staged in LDS (149.5 KB of the 320 KB WGP LDS),
// padded strides make every fragment read bank-conflict-free.
__global__ void __launch_bounds__(256)
sage_gemm_kernel(const float* __restrict__ agg,
                 const float* __restrict__ x,
                 const float* __restrict__ w_l,
                 const float* __restrict__ w_r,
                 const float* __restrict__ b_l,
                 float*       __restrict__ out) {
    __shared__ float wT[K_TOT * WT_STRIDE];   // wT[k][n] = W[n][k]   (129 KB)
    __shared__ float aT[K_TOT * AT_STRIDE];   // aT[k][m] = A[m][k]   (17.4 KB)

    const int mb = blockIdx.x;

    // Stage weights transposed: coalesced global reads, stride-129 LDS rows
    // => consecutive k land in consecutive banks (129 % 64 == 1).
    for (int i = threadIdx.x; i < D_FEAT * D_FEAT; i += 256) {
        int n = i >> 7;
        int k = i & (D_FEAT - 1);
        wT[k * WT_STRIDE + n]            = w_l[i];
        wT[(k + D_FEAT) * WT_STRIDE + n] = w_r[i];
    }

    // Stage this block's 16-row A tile, transposed (shared by all 8 waves).
    const float* aggp = agg + (size_t)mb * 16 * D_FEAT;
    const float* xp   = x   + (size_t)mb * 16 * D_FEAT;
    for (int i = threadIdx.x; i < 16 * D_FEAT; i += 256) {
        int m = i >> 7;
        int k = i & (D_FEAT - 1);
        aT[k * AT_STRIDE + m]            = aggp[i];
        aT[(k + D_FEAT) * AT_STRIDE + m] = xp[i];
    }
    __syncthreads();

    const int lane  = threadIdx.x & 31;
    const int ntile = threadIdx.x >> 5;     // wave id = N tile
    const int mloc  = lane & 15;            // row within A frag / col within B frag
    const int nglb  = ntile * 16 + mloc;    // global output column for this lane
    const int half  = lane >> 4;            // 0: K+{0,1}, 1: K+{2,3} (ISA layout)

    v8f acc = {};
#pragma unroll 4
    for (int kk = 0; kk < K_TOT; kk += 4) {
        const int koff = kk + (half << 1);
        v2f a, b;
        a.x = aT[koff * AT_STRIDE + mloc];        // A[m][koff]
        a.y = aT[(koff + 1) * AT_STRIDE + mloc];  // A[m][koff+1]
        b.x = wT[koff * WT_STRIDE + nglb];        // B[koff][n]
        b.y = wT[(koff + 1) * WT_STRIDE + nglb];  // B[koff+1][n]
        // D = A(16x4) * B(4x16) + C, exact fp32 (RNE), matches reference math.
        acc = __builtin_amdgcn_wmma_f32_16x16x4_f32(
            /*neg_a=*/false, a, /*neg_b=*/false, b,
            /*c_mod=*/(short)0, acc, /*reuse_a=*/false, /*reuse_b=*/false);
    }

    // Epilogue: add bias (same column n for every C VGPR in this lane), store.
    const float bl = b_l[nglb];
    float* outp = out + (size_t)mb * 16 * D_FEAT + nglb;
#pragma unroll
    for (int j = 0; j < 8; ++j) {
        // C/D layout: VGPR j -> row j (lanes 0-15) or row j+8 (lanes 16-31).
        outp[(size_t)(j + half * 8) * D_FEAT] = acc[j] + bl;
    }
}

// ---------------------------------------------------------------------------
extern "C" void kernel_launch(void* const* d_in, const int* in_sizes, int n_in,
                              void* d_out, int out_size, void* d_ws, size_t ws_size,
                              hipStream_t stream) {
    const float* x   = (const float*)d_in[0];
    const int*   row = (const int*)  d_in[1];
    const int*   col = (const int*)  d_in[2];
    const float* ew  = (const float*)d_in[3];
    const float* w_l = (const float*)d_in[4];
    const float* b_l = (const float*)d_in[5];
    const float* w_r = (const float*)d_in[6];
    float*       out = (float*)d_out;

    const int N = in_sizes[0] / D_FEAT;   // 100000 nodes
    const int E = in_sizes[1];            // 3200000 edges

    float* agg = (float*)d_ws;            // N * 128 floats = 51.2 MB scratch

    // 1) agg = 0
    {
        int n4   = N * (D_FEAT / 4);
        int grid = (n4 + 255) / 256;
        sage_zero_kernel<<<grid, 256, 0, stream>>>((float4*)agg, n4);
    }
    // 2) scatter-add SpMM (one wave32 per edge)
    {
        long long threads = (long long)E * 32;
        int grid = (int)((threads + 255) / 256);
        sage_scatter_kernel<<<grid, 256, 0, stream>>>(x, row, col, ew, agg, E);
    }
    // 3) fused WMMA GEMM: out = agg@w_l^T + b_l + x@w_r^T
    {
        int grid = N / 16;                // 6250, exact
        sage_gemm_kernel<<<grid, 256, 0, stream>>>(agg, x, w_l, w_r, b_l, out);
    }
}